// InvConvNear_42090679501036
// MI455X (gfx1250) — compile-verified
//
#include <hip/hip_runtime.h>
#include <hip/hip_bf16.h>

// InvConvNear: b=32, c=192, t=8192, n_split=4, cs=48.
// z[b, ch_i(k), t] = mask[b,t] * sum_j W[i][j] * x[b, ch_j(k), t]
//   ch_j(k) = 96*(j>>1) + 2*k + (j&1)
// logdet[b] = log|det W| * (c/ns) * sum_t mask[b,t]

#define B_  32
#define C_  192
#define T_  8192
#define CS_ 48      // c / n_split
#define TS_ 8       // t-slices per (b,k) row
#define TSLICE (T_ / TS_)   // 1024 t per wave

typedef float v2f __attribute__((ext_vector_type(2)));
typedef float v8f __attribute__((ext_vector_type(8)));

__global__ __launch_bounds__(256) void invconv_wmma_kernel(
    const float* __restrict__ x, const float* __restrict__ xmask,
    const float* __restrict__ w, float* __restrict__ z)
{
    const int gtid = blockIdx.x * blockDim.x + threadIdx.x;
    const int wid  = gtid >> 5;        // global wave id
    const int lane = gtid & 31;
    const int n    = lane & 15;        // column / matrix row index
    const int hi   = lane >> 4;        // half-wave select

    const int b   = wid / (CS_ * TS_);
    const int rem = wid % (CS_ * TS_);
    const int k   = rem / TS_;
    const int ts  = rem % TS_;

    // ---- A matrix (16x4 f32): rows 0-3 and 8-11 hold W, others zero.
    // Layout: lanes 0-15 -> {K0,K1} in {v0,v1}; lanes 16-31 -> {K2,K3}.
    v2f A;
    if ((n & 4) == 0) {
        const int wr = n & 3;
        A.x = w[wr * 4 + 2 * hi + 0];
        A.y = w[wr * 4 + 2 * hi + 1];
    } else {
        A.x = 0.0f; A.y = 0.0f;
    }

    // Input/output channel rows for this k: {2k, 2k+1, 96+2k, 97+2k}
    const size_t rowlo = ((size_t)b * C_ + 2 * k) * T_;       // ch 2k
    const size_t rowhi = ((size_t)b * C_ + 96 + 2 * k) * T_;  // ch 96+2k
    // lanes 0-15 read channel pair (2k, 2k+1); lanes 16-31 read (96+2k, 97+2k)
    const float* bp = x + (hi ? rowhi : rowlo);
    const float* mp = xmask + (size_t)b * T_;

    float* zp0 = z + rowlo;        // i=0 -> ch 2k
    float* zp1 = z + rowlo + T_;   // i=1 -> ch 2k+1
    float* zp2 = z + rowhi;        // i=2 -> ch 96+2k
    float* zp3 = z + rowhi + T_;   // i=3 -> ch 97+2k

    const v8f czero = {};

    int t0 = ts * TSLICE;
    const int tend = t0 + TSLICE;
    #pragma unroll 2
    for (; t0 < tend; t0 += 32) {
        // B matrix (4x16 f32): VGPR0 = K0 (lanes 0-15) | K2 (lanes 16-31),
        //                      VGPR1 = K1 (lanes 0-15) | K3 (lanes 16-31).
        v2f B1, B2;
        B1.x = __builtin_nontemporal_load(bp + t0 + n);            // K0/K2, t0..t0+15
        B1.y = __builtin_nontemporal_load(bp + T_ + t0 + n);       // K1/K3
        B2.x = __builtin_nontemporal_load(bp + t0 + 16 + n);       // t0+16..t0+31
        B2.y = __builtin_nontemporal_load(bp + T_ + t0 + 16 + n);
        const float m = mp[t0 + lane];                             // cached (reused by 48 k-waves)

        // D rows 0-3 (lanes 0-15) and duplicated rows 8-11 (lanes 16-31)
        v8f d1 = __builtin_amdgcn_wmma_f32_16x16x4_f32(
            false, A, false, B1, (short)0, czero, false, false);
        v8f d2 = __builtin_amdgcn_wmma_f32_16x16x4_f32(
            false, A, false, B2, (short)0, czero, false, false);

        // lane L < 16: z_i(t0+L) in d1[i];  lane L >= 16: z_i(t0+L) in d2[i]
        const float o0 = (hi ? d2[0] : d1[0]) * m;
        const float o1 = (hi ? d2[1] : d1[1]) * m;
        const float o2 = (hi ? d2[2] : d1[2]) * m;
        const float o3 = (hi ? d2[3] : d1[3]) * m;

        __builtin_nontemporal_store(o0, zp0 + t0 + lane);
        __builtin_nontemporal_store(o1, zp1 + t0 + lane);
        __builtin_nontemporal_store(o2, zp2 + t0 + lane);
        __builtin_nontemporal_store(o3, zp3 + t0 + lane);
    }
}

__device__ __forceinline__ float det3(float a, float b, float c,
                                      float d, float e, float f,
                                      float g, float h, float i)
{
    return a * (e * i - f * h) - b * (d * i - f * g) + c * (d * h - e * g);
}

__global__ __launch_bounds__(256) void invconv_logdet_kernel(
    const float* __restrict__ xmask, const float* __restrict__ w,
    float* __restrict__ logdet)
{
    __shared__ float sred[256];
    const int b   = blockIdx.x;
    const int tid = threadIdx.x;

    const float* mp = xmask + (size_t)b * T_;
    float s = 0.0f;
    for (int t = tid; t < T_; t += 256) s += mp[t];
    sred[tid] = s;
    __syncthreads();
    for (int off = 128; off > 0; off >>= 1) {
        if (tid < off) sred[tid] += sred[tid + off];
        __syncthreads();
    }

    if (tid == 0) {
        // 4x4 determinant by cofactor expansion along row 0
        const float det =
              w[0] * det3(w[5],  w[6],  w[7],  w[9],  w[10], w[11], w[13], w[14], w[15])
            - w[1] * det3(w[4],  w[6],  w[7],  w[8],  w[10], w[11], w[12], w[14], w[15])
            + w[2] * det3(w[4],  w[5],  w[7],  w[8],  w[9],  w[11], w[12], w[13], w[15])
            - w[3] * det3(w[4],  w[5],  w[6],  w[8],  w[9],  w[10], w[12], w[13], w[14]);
        logdet[b] = logf(fabsf(det)) * ((float)C_ / 4.0f) * sred[0];
    }
}

extern "C" void kernel_launch(void* const* d_in, const int* in_sizes, int n_in,
                              void* d_out, int out_size, void* d_ws, size_t ws_size,
                              hipStream_t stream) {
    const float* x     = (const float*)d_in[0];   // (32,192,8192) f32
    const float* xmask = (const float*)d_in[1];   // (32,1,8192)   f32
    const float* w     = (const float*)d_in[2];   // (4,4)         f32
    float* z      = (float*)d_out;                          // 50331648 floats
    float* logdet = z + (size_t)B_ * C_ * T_;               // +32 floats

    // 32 * 48 * 8 waves of 32 threads -> 393216 threads -> 1536 blocks of 256
    const int total_threads = B_ * CS_ * TS_ * 32;
    invconv_wmma_kernel<<<total_threads / 256, 256, 0, stream>>>(x, xmask, w, z);
    invconv_logdet_kernel<<<B_, 256, 0, stream>>>(xmask, w, logdet);
}